// ROIAlignRotated_27178553049519
// MI455X (gfx1250) — compile-verified
//
#include <hip/hip_runtime.h>

// ---------------------------------------------------------------------------
// RoI Align Rotated (fwd), f32. feats:(2,256,200,304)  rois:(1000,6)
// out:(1000,256,7,7).  Memory/gather bound; feature map (124.5MB) fits MI455X
// 192MB L2, so default RT temporal hints keep it resident.
// ---------------------------------------------------------------------------

namespace {
constexpr int kC     = 256;
constexpr int kH     = 200;
constexpr int kW     = 304;
constexpr int kOutHW = 7;
constexpr int kSR    = 2;
constexpr int kSamp  = kOutHW * kSR;     // 14 sample rows/cols
constexpr int kNS    = kSamp * kSamp;    // 196 samples per ROI
constexpr int kBins  = kOutHW * kOutHW;  // 49 output bins per (roi, channel)
constexpr float kScale = 0.25f;
constexpr float kPiOver180 = 3.14159265358979323846f / 180.0f;
}

// gfx1250 async global<->LDS paths (guarded; plain fallbacks keep the build
// safe on toolchains lacking the builtins).
#if defined(__HIP_DEVICE_COMPILE__) && defined(__gfx1250__) &&                 \
    __has_builtin(__builtin_amdgcn_global_load_async_to_lds_b32) &&            \
    __has_builtin(__builtin_amdgcn_s_wait_asynccnt)
#define ROI_ASYNC_LDS 1
typedef __attribute__((address_space(1))) int GlobalInt;
typedef __attribute__((address_space(3))) int LdsInt;
#else
#define ROI_ASYNC_LDS 0
#endif

#if ROI_ASYNC_LDS && __has_builtin(__builtin_amdgcn_global_store_async_from_lds_b128)
#define ROI_ASYNC_STORE 1
typedef int v4i __attribute__((ext_vector_type(4)));
typedef __attribute__((address_space(1))) v4i GlobalV4i;
typedef __attribute__((address_space(3))) v4i LdsV4i;
#else
#define ROI_ASYNC_STORE 0
#endif

extern "C" __global__ __launch_bounds__(kC)
void roi_align_rotated_f32(const float* __restrict__ feats,
                           const float* __restrict__ rois,
                           float* __restrict__ out)
{
    __shared__ float  s_roi[8];          // staged ROI row (6 used)
    __shared__ int4   s_off[kNS];        // 4 corner offsets per sample
    __shared__ float4 s_w[kNS];          // 4 bilinear weights per sample
    __shared__ __align__(16) float s_tile[kC * kBins]; // (c,bin) output tile

    const int n   = blockIdx.x;
    const int tid = threadIdx.x;
    const float* __restrict__ r = rois + (size_t)n * 6;

    // ---- stage ROI row into LDS (gfx1250 async global->LDS) ----------------
#if ROI_ASYNC_LDS
    if (tid < 6) {
        __builtin_amdgcn_global_load_async_to_lds_b32(
            (GlobalInt*)(r + tid), (LdsInt*)(&s_roi[tid]), 0, 0);
    }
    __builtin_amdgcn_s_wait_asynccnt(0);
#else
    if (tid < 6) s_roi[tid] = r[tid];
#endif
    __syncthreads();

    // ---- phase 1: per-sample descriptors (196 threads) ---------------------
    if (tid < kNS) {
        const float cw = s_roi[1] * kScale - 0.5f;
        const float ch = s_roi[2] * kScale - 0.5f;
        const float rw = s_roi[3] * kScale;
        const float rh = s_roi[4] * kScale;
        const float th = s_roi[5] * kPiOver180;
        const float c_t = cosf(th);
        const float s_t = sinf(th);

        const int iy = tid / kSamp;
        const int ix = tid - iy * kSamp;

        const float bin_h = rh * (1.0f / kOutHW);
        const float bin_w = rw * (1.0f / kOutHW);
        const float ty = ((float)iy + 0.5f) * (1.0f / kSR);
        const float tx = ((float)ix + 0.5f) * (1.0f / kSR);
        const float yy = -0.5f * rh + ty * bin_h;
        const float xx = -0.5f * rw + tx * bin_w;

        float y = yy * c_t - xx * s_t + ch;
        float x = yy * s_t + xx * c_t + cw;

        const bool empty = (y < -1.0f) | (y > (float)kH) |
                           (x < -1.0f) | (x > (float)kW);

        y = fmaxf(y, 0.0f);
        x = fmaxf(x, 0.0f);
        const int yl = min((int)y, kH - 1);   // y >= 0 -> trunc == floor
        const int xl = min((int)x, kW - 1);
        const int yh = min(yl + 1, kH - 1);
        const int xh = min(xl + 1, kW - 1);

        const float ly = fminf(fmaxf(y - (float)yl, 0.0f), 1.0f);
        const float lx = fminf(fmaxf(x - (float)xl, 0.0f), 1.0f);
        const float hy = 1.0f - ly, hx = 1.0f - lx;

        float w1 = hy * hx, w2 = hy * lx, w3 = ly * hx, w4 = ly * lx;
        if (empty) { w1 = w2 = w3 = w4 = 0.0f; }

        s_off[tid] = make_int4(yl * kW + xl, yl * kW + xh,
                               yh * kW + xl, yh * kW + xh);
        s_w[tid]   = make_float4(w1, w2, w3, w4);
    }
    __syncthreads();

    // ---- phase 2: lane = channel, gather + accumulate ----------------------
    // 32-bit per-lane element offset (tensor is 31.1M elements < 2^31 bytes)
    // keeps each gather to one v_add + saddr-based global_load.
    const int b = (int)s_roi[0];
    const int cbase = (b * kC + tid) * (kH * kW);

    // Warm the first gather lines (gfx1250 global_prefetch_b8).
    __builtin_prefetch((const void*)(feats + cbase + s_off[0].x), 0, 3);
    __builtin_prefetch((const void*)(feats + cbase + s_off[kNS - 1].w), 0, 3);

    for (int bin = 0; bin < kBins; ++bin) {
        const int ph = bin / kOutHW;
        const int pw = bin - ph * kOutHW;
        const int s0 = (ph * kSR) * kSamp + pw * kSR;

        float acc = 0.0f;
#pragma unroll
        for (int k = 0; k < 4; ++k) {
            const int s = s0 + (k >> 1) * kSamp + (k & 1);
            const int4   o = s_off[s];   // ds_load_b128, broadcast (no conflicts)
            const float4 w = s_w[s];
            acc = fmaf(w.x, feats[cbase + o.x], acc);
            acc = fmaf(w.y, feats[cbase + o.y], acc);
            acc = fmaf(w.z, feats[cbase + o.z], acc);
            acc = fmaf(w.w, feats[cbase + o.w], acc);
        }
        s_tile[tid * kBins + bin] = acc * (1.0f / (kSR * kSR));
    }
    __syncthreads();

    // ---- coalesced writeback: out[n] is contiguous (c,ph,pw) ---------------
    float* __restrict__ on = out + (size_t)n * (kC * kBins);
#if ROI_ASYNC_STORE
    // LDS -> global directly (no VGPR round trip), 16B per lane per op.
    for (int i = tid; i < (kC * kBins) / 4; i += kC) {
        __builtin_amdgcn_global_store_async_from_lds_b128(
            (GlobalV4i*)(on + 4 * i), (LdsV4i*)(&s_tile[4 * i]), 0, 0);
    }
    // s_endpgm performs an implicit wait-idle, covering outstanding ASYNCcnt.
#else
    const float4* __restrict__ t4 = (const float4*)s_tile;
    float4* __restrict__ o4 = (float4*)on;
    for (int i = tid; i < (kC * kBins) / 4; i += kC) o4[i] = t4[i];
#endif
}

extern "C" void kernel_launch(void* const* d_in, const int* in_sizes, int n_in,
                              void* d_out, int out_size, void* d_ws, size_t ws_size,
                              hipStream_t stream) {
    const float* feats = (const float*)d_in[0];
    const float* rois  = (const float*)d_in[1];
    float* out = (float*)d_out;
    const int N = in_sizes[1] / 6;   // 1000 ROIs
    roi_align_rotated_f32<<<dim3(N), dim3(kC), 0, stream>>>(feats, rois, out);
}